// MatrixKANLinear_65712999629032
// MI455X (gfx1250) — compile-verified
//
#include <hip/hip_runtime.h>

typedef __attribute__((ext_vector_type(2))) float v2f;
typedef __attribute__((ext_vector_type(8))) float v8f;

#define NSAMP 2048
#define FINC  128
#define FOUTC 128

// Workspace layout (float element offsets):
//  silu   : [0,       262144)           silu(x), A-matrix of base GEMM
//  bspl4  : [262144,  1310720)          4 planes [J][n][a] of basis values
//  cfl    : [1310720, 1572864)          int segment index per (n,a)
//  CPt    : [1572864, 1703936)          4 * (256 x 128) spline weights, Bt layout
//  T      : [1703936, 3801088)          4 * (2048 x 256) spline GEMM outputs
//  baseO  : [3801088, 4063232)          base GEMM output
// total ~16.3 MB

__global__ void kan_prep(const float* __restrict__ x, float* __restrict__ siluA,
                         float* __restrict__ bspl4, int* __restrict__ cfl) {
  int gid = blockIdx.x * blockDim.x + threadIdx.x;   // n*128 + a
  float v = x[gid];
  float sg = 1.0f / (1.0f + __expf(-v));
  siluA[gid] = v * sg;

  float xn = fminf(fmaxf(v, -1.0f), 1.0f);
  float t  = (xn + 2.2f) / 0.4f;        // (xn - G0)/H in fp32, like the reference
  float fl = floorf(t);
  float u  = t - fl;                     // u from UNclamped floor (matches ref)
  int c = (int)fl;
  c = (c < 3) ? 3 : c;
  c = (c > 7) ? 7 : c;
  c -= 3;                                // segment in [0,4]

  float u2 = u * u, u3 = u2 * u;
  const float k6 = 1.0f / 6.0f;
  float b0 = (1.0f - 3.0f*u + 3.0f*u2 - u3) * k6;
  float b1 = (4.0f - 6.0f*u2 + 3.0f*u3) * k6;
  float b2 = (1.0f + 3.0f*u + 3.0f*u2 - 3.0f*u3) * k6;
  float b3 = u3 * k6;

  bspl4[0 * NSAMP * FINC + gid] = b0;
  bspl4[1 * NSAMP * FINC + gid] = b1;
  bspl4[2 * NSAMP * FINC + gid] = b2;
  bspl4[3 * NSAMP * FINC + gid] = b3;
  cfl[gid] = c;
}

// CPt[j*256*128 + (p*8+g)*128 + a] = sw[(a*128 + 32*j + p)*8 + g]
__global__ void kan_wprep(const float* __restrict__ sw, float* __restrict__ CPt) {
  int gid = blockIdx.x * blockDim.x + threadIdx.x;   // 4*256*128 = 131072
  int j = gid >> 15;
  int r = gid & 32767;
  int col = r >> 7;          // p*8 + g
  int a   = r & 127;
  int p = col >> 3, g = col & 7;
  CPt[gid] = sw[(a * FINC + 32 * j + p) * 8 + g];
}

// C[M x N] = A[M x K] @ Bt[N x K]^T  (Bt stored row-major as N rows of K)
// One wave per 16x16 output tile, fp32 WMMA, K-step 4.
// A fragment (16x4 f32): lanes 0-15 hold M=l, VGPR0=K, VGPR1=K+1;
//                        lanes 16-31 hold M=l, VGPR0=K+2, VGPR1=K+3.
// B fragment mirrors with N=lane.  C/D: VGPR r -> (M = r + 8*half, N = l).
__global__ void gemm_wmma(const float* __restrict__ A, const float* __restrict__ Bt,
                          float* __restrict__ C, int N, int K, int ntShift,
                          long sA, long sB, long sC) {
  const float* Ab = A + (size_t)blockIdx.y * sA;
  const float* Bb = Bt + (size_t)blockIdx.y * sB;
  float*       Cb = C + (size_t)blockIdx.y * sC;

  int wave = (blockIdx.x * blockDim.x + threadIdx.x) >> 5;
  int lane = threadIdx.x & 31;
  int tileN = wave & ((1 << ntShift) - 1);
  int tileM = wave >> ntShift;
  int half = lane >> 4;      // which K-pair this half-wave holds
  int l    = lane & 15;

  const float* arow = Ab + (size_t)(tileM * 16 + l) * K + 2 * half;
  const float* brow = Bb + (size_t)(tileN * 16 + l) * K + 2 * half;

  v8f acc = {};
#pragma unroll 8
  for (int k = 0; k < K; k += 4) {
    v2f a = *(const v2f*)(arow + k);
    v2f b = *(const v2f*)(brow + k);
    // D = A(16x4) x B(4x16) + C, pure fp32 (matches reference precision)
    acc = __builtin_amdgcn_wmma_f32_16x16x4_f32(false, a, false, b,
                                                (short)0, acc, false, false);
  }

  int col = tileN * 16 + l;
  float* out = Cb + (size_t)(tileM * 16 + half * 8) * N + col;
#pragma unroll
  for (int r = 0; r < 8; ++r) out[(size_t)r * N] = acc[r];
}

// out[n, 4p+q] = baseO[n,4p+q] + sum_J T[J][n][p*8 + cfl[n,32J+p] + q]
__global__ void kan_epi(const float* __restrict__ baseO, const float* __restrict__ T,
                        const int* __restrict__ cfl, float* __restrict__ out) {
  int gid = blockIdx.x * blockDim.x + threadIdx.x;   // n*128 + b
  int n = gid >> 7;
  int b = gid & 127;
  int p = b >> 2, q = b & 3;
  float acc = baseO[gid];
#pragma unroll
  for (int J = 0; J < 4; ++J) {
    int c = cfl[(n << 7) + 32 * J + p];
    acc += T[((size_t)J * NSAMP + n) * 256 + p * 8 + c + q];
  }
  out[gid] = acc;
}

extern "C" void kernel_launch(void* const* d_in, const int* in_sizes, int n_in,
                              void* d_out, int out_size, void* d_ws, size_t ws_size,
                              hipStream_t stream) {
  (void)in_sizes; (void)n_in; (void)out_size; (void)ws_size;
  const float* x  = (const float*)d_in[0];   // (2048, 128)
  const float* bw = (const float*)d_in[1];   // (128, 128) row-major [o][i]
  const float* sw = (const float*)d_in[2];   // (128, 128, 8) flat

  float* ws    = (float*)d_ws;
  float* siluA = ws;
  float* bspl4 = ws + 262144;
  int*   cfl   = (int*)(ws + 1310720);
  float* CPt   = ws + 1572864;
  float* T     = ws + 1703936;
  float* baseO = ws + 3801088;
  float* out   = (float*)d_out;

  kan_prep <<<1024, 256, 0, stream>>>(x, siluA, bspl4, cfl);
  kan_wprep<<<512, 256, 0, stream>>>(sw, CPt);

  // base GEMM: M=2048, N=128, K=128 -> 1024 waves -> 128 blocks of 8 waves
  gemm_wmma<<<dim3(128, 1), 256, 0, stream>>>(siluA, bw, baseO,
                                              128, 128, /*ntShift=*/3,
                                              0, 0, 0);
  // spline GEMMs: 4 batches of M=2048, N=256, K=128 -> 2048 waves/batch
  gemm_wmma<<<dim3(256, 4), 256, 0, stream>>>(bspl4, CPt, T,
                                              256, 128, /*ntShift=*/4,
                                              (long)NSAMP * FINC,
                                              (long)256 * 128,
                                              (long)NSAMP * 256);

  kan_epi<<<1024, 256, 0, stream>>>(baseO, T, cfl, out);
}